// s2v_embedding_35227321762171
// MI455X (gfx1250) — compile-verified
//
#include <hip/hip_runtime.h>

typedef __attribute__((ext_vector_type(2))) float v2f;
typedef __attribute__((ext_vector_type(8))) float v8f;

#define F_DIM 128
#define D_DIM 64
#define T_ITERS 10

// ---------------------------------------------------------------------------
// helpers
// ---------------------------------------------------------------------------
__device__ __forceinline__ void atomic_add_f32(float* p, float v) {
    // lowers to global_atomic_add_f32 (hardware f32 atomic at L2)
    unsafeAtomicAdd(p, v);
}

__global__ __launch_bounds__(256) void zero_kernel(float* __restrict__ p, int n) {
    int i = blockIdx.x * 256 + threadIdx.x;
    if (i < n) p[i] = 0.0f;
}

// ---------------------------------------------------------------------------
// Wx = x @ W1^T + b1     x:[N,128]  W1:[64,128]  -> Wx:[N,64]
// one wave = one 16x16 output tile; 4 waves/block cover the 64 output dims
// ---------------------------------------------------------------------------
__global__ __launch_bounds__(128) void wx_kernel(const float* __restrict__ x,
                                                 const float* __restrict__ W1,
                                                 const float* __restrict__ b1,
                                                 float* __restrict__ Wx, int N) {
    const int lane  = threadIdx.x & 31;
    const int tileN = threadIdx.x >> 5;   // 0..3 -> output-dim tile
    const int tileM = blockIdx.x;         // node-row tile
    const int lid   = lane & 15;
    const int khalf = lane >> 4;          // 0/1 -> +2 K offset (A/B f32 layout)

    int rowA = tileM * 16 + lid;
    if (rowA >= N) rowA = N - 1;          // clamp (no divergence before WMMA)
    const int colB = tileN * 16 + lid;    // W1 row = output dim

    const float* aptr = x  + (size_t)rowA * F_DIM + khalf * 2;
    const float* bptr = W1 + (size_t)colB * F_DIM + khalf * 2;

    v8f c = {0.f, 0.f, 0.f, 0.f, 0.f, 0.f, 0.f, 0.f};
#pragma unroll
    for (int k0 = 0; k0 < F_DIM; k0 += 4) {
        v2f a = *(const v2f*)(aptr + k0);
        v2f b = *(const v2f*)(bptr + k0);
        c = __builtin_amdgcn_wmma_f32_16x16x4_f32(false, a, false, b,
                                                  (short)0, c, false, false);
    }

    const int n     = lid;                // C/D: col = lane&15
    const int mbase = khalf * 8;          // C/D: rows 0-7 (lanes 0-15) / 8-15
    const float bias = b1[tileN * 16 + n];
#pragma unroll
    for (int r = 0; r < 8; ++r) {
        int row = tileM * 16 + mbase + r;
        if (row < N)
            Wx[(size_t)row * D_DIM + tileN * 16 + n] = c[r] + bias;
    }
}

// ---------------------------------------------------------------------------
// iteration 0:  emb = relu(Wx + b2)   (agg of zero embedding is zero)
// ---------------------------------------------------------------------------
__global__ __launch_bounds__(256) void relu0_kernel(const float* __restrict__ Wx,
                                                    const float* __restrict__ b2,
                                                    float* __restrict__ emb, int n) {
    int i = blockIdx.x * 256 + threadIdx.x;
    if (i < n) emb[i] = fmaxf(Wx[i] + b2[i & (D_DIM - 1)], 0.0f);
}

// ---------------------------------------------------------------------------
// agg[dst] += emb[src]  (64 consecutive threads per edge -> coalesced rows)
// emb/agg are L2-resident (12.8 MB each)
// ---------------------------------------------------------------------------
__global__ __launch_bounds__(256) void agg_kernel(const float* __restrict__ emb,
                                                  const int* __restrict__ esrc,
                                                  const int* __restrict__ edst,
                                                  float* __restrict__ agg, int E) {
    int tid = blockIdx.x * 256 + threadIdx.x;
    int e   = tid >> 6;
    if (e >= E) return;
    int d = tid & 63;
    int s = esrc[e];
    int t = edst[e];
    float v = emb[(size_t)s * D_DIM + d];
    atomic_add_f32(&agg[(size_t)t * D_DIM + d], v);
}

// ---------------------------------------------------------------------------
// emb_out = relu(Wx + agg @ W2^T + b2)    agg:[N,64] W2:[64,64]
// ---------------------------------------------------------------------------
__global__ __launch_bounds__(128) void update_kernel(const float* __restrict__ agg,
                                                     const float* __restrict__ W2,
                                                     const float* __restrict__ b2,
                                                     const float* __restrict__ Wx,
                                                     float* __restrict__ emb_out,
                                                     int N) {
    const int lane  = threadIdx.x & 31;
    const int tileN = threadIdx.x >> 5;
    const int tileM = blockIdx.x;
    const int lid   = lane & 15;
    const int khalf = lane >> 4;

    int rowA = tileM * 16 + lid;
    if (rowA >= N) rowA = N - 1;
    const int colB = tileN * 16 + lid;

    const float* aptr = agg + (size_t)rowA * D_DIM + khalf * 2;
    const float* bptr = W2  + (size_t)colB * D_DIM + khalf * 2;

    v8f c = {0.f, 0.f, 0.f, 0.f, 0.f, 0.f, 0.f, 0.f};
#pragma unroll
    for (int k0 = 0; k0 < D_DIM; k0 += 4) {
        v2f a = *(const v2f*)(aptr + k0);
        v2f b = *(const v2f*)(bptr + k0);
        c = __builtin_amdgcn_wmma_f32_16x16x4_f32(false, a, false, b,
                                                  (short)0, c, false, false);
    }

    const int n     = lid;
    const int mbase = khalf * 8;
    const float bias = b2[tileN * 16 + n];
#pragma unroll
    for (int r = 0; r < 8; ++r) {
        int row = tileM * 16 + mbase + r;
        if (row < N) {
            size_t idx = (size_t)row * D_DIM + tileN * 16 + n;
            emb_out[idx] = fmaxf(c[r] + Wx[idx] + bias, 0.0f);
        }
    }
}

// ---------------------------------------------------------------------------
// deterministic two-stage column reduction: out[d] = sum_n emb[n][d]
// ---------------------------------------------------------------------------
__global__ __launch_bounds__(256) void reduce1_kernel(const float* __restrict__ emb,
                                                      float* __restrict__ partial,
                                                      int N) {
    __shared__ float lds[256];
    const int d   = threadIdx.x & 63;
    const int sub = threadIdx.x >> 6;      // 0..3
    float local = 0.f;
    for (int r = blockIdx.x * 4 + sub; r < N; r += gridDim.x * 4)
        local += emb[(size_t)r * D_DIM + d];
    lds[threadIdx.x] = local;
    __syncthreads();
    if (sub == 0)
        partial[blockIdx.x * D_DIM + d] =
            lds[d] + lds[64 + d] + lds[128 + d] + lds[192 + d];
}

__global__ __launch_bounds__(64) void reduce2_kernel(const float* __restrict__ partial,
                                                     float* __restrict__ out,
                                                     int nblocks) {
    int d = threadIdx.x;                   // 64 threads
    float s = 0.f;
    for (int b = 0; b < nblocks; ++b) s += partial[b * D_DIM + d];
    out[d] = s;
}

// ---------------------------------------------------------------------------
// launch
// ---------------------------------------------------------------------------
extern "C" void kernel_launch(void* const* d_in, const int* in_sizes, int n_in,
                              void* d_out, int out_size, void* d_ws, size_t ws_size,
                              hipStream_t stream) {
    const float* x    = (const float*)d_in[0];
    const float* W1   = (const float*)d_in[1];
    const float* b1   = (const float*)d_in[2];
    const float* W2   = (const float*)d_in[3];
    const float* b2   = (const float*)d_in[4];
    const int*   esrc = (const int*)d_in[5];
    const int*   edst = (const int*)d_in[6];

    const int N = in_sizes[0] / F_DIM;
    const int E = in_sizes[5];
    float* out = (float*)d_out;

    // workspace layout (floats)
    float* ws = (float*)d_ws;
    const size_t nd = (size_t)N * D_DIM;
    float* Wx      = ws;
    float* emb_a   = Wx + nd;
    float* emb_b   = emb_a + nd;
    float* agg     = emb_b + nd;
    float* partial = agg + nd;

    const int tilesM     = (N + 15) / 16;
    const int elemBlocks = (int)((nd + 255) / 256);
    const int aggBlocks  = (int)(((size_t)E * D_DIM + 255) / 256);
    const int RB         = 512;

    // Wx = x@W1^T + b1 (loop-invariant)
    wx_kernel<<<tilesM, 128, 0, stream>>>(x, W1, b1, Wx, N);

    // iteration 1 of T: agg(emb0=0)=0  ->  emb = relu(Wx + b2)
    relu0_kernel<<<elemBlocks, 256, 0, stream>>>(Wx, b2, emb_a, (int)nd);

    float* cur = emb_a;
    float* nxt = emb_b;
    for (int t = 1; t < T_ITERS; ++t) {
        zero_kernel<<<elemBlocks, 256, 0, stream>>>(agg, (int)nd);
        agg_kernel<<<aggBlocks, 256, 0, stream>>>(cur, esrc, edst, agg, E);
        update_kernel<<<tilesM, 128, 0, stream>>>(agg, W2, b2, Wx, nxt, N);
        float* tmp = cur; cur = nxt; nxt = tmp;
    }

    // out[d] = sum over nodes (deterministic tree)
    reduce1_kernel<<<RB, 256, 0, stream>>>(cur, partial, N);
    reduce2_kernel<<<1, 64, 0, stream>>>(partial, out, RB);
}